// GNNLayer_11905649345047
// MI455X (gfx1250) — compile-verified
//
#include <hip/hip_runtime.h>
#include <stdint.h>

#define NNODES 50000
#define NEDGES 800000
#define ETOT   (NEDGES + NNODES)   // 850000 (self loops appended)

typedef float v2f __attribute__((ext_vector_type(2)));
typedef float v8f __attribute__((ext_vector_type(8)));

// ---------- monotone float <-> uint key (for native u32 atomicMax segment-max) ----------
__device__ __forceinline__ unsigned flipF(float f) {
    unsigned u = __float_as_uint(f);
    unsigned mask = (unsigned)(-(int)(u >> 31)) | 0x80000000u;
    return u ^ mask;
}
__device__ __forceinline__ float unflipF(unsigned u) {
    unsigned mask = ((u >> 31) - 1u) | 0x80000000u;
    return __uint_as_float(u ^ mask);
}

__device__ __forceinline__ void edge_sd(const int* __restrict__ ei, int e, int& s, int& d) {
    if (e < NEDGES) { s = ei[e]; d = ei[NEDGES + e]; }
    else            { s = e - NEDGES; d = s; }          // self loop
}

// ---------------------------------------------------------------- zero fill
__global__ void zero_f32(float* __restrict__ p, long n) {
    long i = (long)blockIdx.x * blockDim.x + threadIdx.x;
    long st = (long)gridDim.x * blockDim.x;
    for (; i < n; i += st) p[i] = 0.0f;
}

// ---------------------------------------------------------------- fp32 WMMA GEMM
// C[M,N] = A[M,K] @ B[K,N], all row-major. One wave per 16x64 strip (4 N-tiles
// share the A fragment). Uses V_WMMA_F32_16X16X4_F32 (wave32).
__global__ __launch_bounds__(256)
void gemm_f32_wmma(const float* __restrict__ A, const float* __restrict__ B,
                   float* __restrict__ C, int M, int N, int K) {
    const int wave = (blockIdx.x * blockDim.x + threadIdx.x) >> 5;
    const int lane = threadIdx.x & 31;
    const int stripsN = N >> 6;               // strips of 64 columns
    const int tm = wave / stripsN;
    const int tn = (wave % stripsN) * 4;      // first 16-wide N tile of this strip
    if (tm * 16 >= M) return;

    const int m   = lane & 15;                // row within tile (both halves)
    const int kh  = (lane >> 4) << 1;         // K offset of this half-wave: 0 or 2
    const int n0  = lane & 15;                // column within tile for B/C

    v8f acc[4];
    #pragma unroll
    for (int t = 0; t < 4; ++t) acc[t] = (v8f){0.f,0.f,0.f,0.f,0.f,0.f,0.f,0.f};

    const float* __restrict__ Arow = A + (long)(tm * 16 + m) * K;
    for (int k0 = 0; k0 < K; k0 += 4) {
        v2f a;                                // A 16x4 frag: VGPR j -> K = k0+kh+j
        a.x = Arow[k0 + kh + 0];
        a.y = Arow[k0 + kh + 1];
        #pragma unroll
        for (int t = 0; t < 4; ++t) {
            const int col = (tn + t) * 16 + n0;
            v2f b;                            // B 4x16 frag: VGPR j -> K = k0+kh+j
            b.x = B[(long)(k0 + kh + 0) * N + col];
            b.y = B[(long)(k0 + kh + 1) * N + col];
            acc[t] = __builtin_amdgcn_wmma_f32_16x16x4_f32(
                         false, a, false, b, (short)0, acc[t], false, false);
        }
    }
    // C/D layout: VGPR v -> M = v + 8*(lane>=16), N = lane&15
    const int moff = (lane >> 4) * 8;
    #pragma unroll
    for (int t = 0; t < 4; ++t) {
        #pragma unroll
        for (int v = 0; v < 8; ++v) {
            C[(long)(tm * 16 + v + moff) * N + (tn + t) * 16 + n0] = acc[t][v];
        }
    }
}

// ---------------------------------------------------------------- a_src/a_dst dots
__global__ void attn_scores(const float* __restrict__ xh,
                            const float* __restrict__ att_src,
                            const float* __restrict__ att_dst,
                            float* __restrict__ asrc, float* __restrict__ adst,
                            int H, int C) {
    int idx = blockIdx.x * blockDim.x + threadIdx.x;   // node*H + h
    if (idx >= NNODES * H) return;
    int n = idx / H, h = idx % H;
    const float* x = xh + (long)n * H * C + (long)h * C;
    const float* as = att_src + h * C;
    const float* ad = att_dst + h * C;
    float ss = 0.f, sd = 0.f;
    for (int c = 0; c < C; ++c) { float v = x[c]; ss += v * as[c]; sd += v * ad[c]; }
    asrc[idx] = ss;
    adst[idx] = sd;
}

// ---------------------------------------------------------------- per-edge score + segment max
__global__ void edge_max(const int* __restrict__ ei,
                         const float* __restrict__ asrc, const float* __restrict__ adst,
                         float* __restrict__ e_out, unsigned* __restrict__ emax, int H) {
    int idx = blockIdx.x * blockDim.x + threadIdx.x;   // edge*H + h
    if (idx >= ETOT * H) return;
    int e = idx / H, h = idx % H;
    int s, d; edge_sd(ei, e, s, d);
    float x = asrc[s * H + h] + adst[d * H + h];
    x = (x > 0.f) ? x : 0.2f * x;                      // leaky_relu(., 0.2)
    e_out[idx] = x;
    atomicMax(&emax[d * H + h], flipF(x));             // GLOBAL_ATOMIC_MAX_U32
}

// ---------------------------------------------------------------- exp + segment sum
__global__ void edge_expsum(const int* __restrict__ ei,
                            float* __restrict__ e_io, const unsigned* __restrict__ emax,
                            float* __restrict__ esum, int H) {
    int idx = blockIdx.x * blockDim.x + threadIdx.x;
    if (idx >= ETOT * H) return;
    int e = idx / H, h = idx % H;
    int s, d; edge_sd(ei, e, s, d); (void)s;
    float m  = unflipF(emax[d * H + h]);
    float ex = expf(e_io[idx] - m);
    e_io[idx] = ex;
    atomicAdd(&esum[d * H + h], ex);                   // GLOBAL_ATOMIC_ADD_F32
}

// ---------------------------------------------------------------- attention-weighted scatter-add
// one wave32 per edge; float4 coalesced gathers, native f32 atomics scatter.
__global__ __launch_bounds__(256)
void edge_aggregate(const int* __restrict__ ei, const float* __restrict__ ex,
                    const float* __restrict__ esum, const float* __restrict__ xh,
                    float* __restrict__ out, int H, int C) {
    int gid  = blockIdx.x * blockDim.x + threadIdx.x;
    int e    = gid >> 5;
    int lane = gid & 31;
    if (e >= ETOT) return;
    int s, d; edge_sd(ei, e, s, d);
    const int HC = H * C;
    const float* __restrict__ xs = xh + (long)s * HC;
    float* __restrict__ od = out + (long)d * HC;
    for (int base = lane * 4; base < HC; base += 128) {
        int h = base / C;
        float coeff = ex[(long)e * H + h] / esum[d * H + h];
        float4 v = *(const float4*)(xs + base);
        atomicAdd(od + base + 0, coeff * v.x);
        atomicAdd(od + base + 1, coeff * v.y);
        atomicAdd(od + base + 2, coeff * v.z);
        atomicAdd(od + base + 3, coeff * v.w);
    }
}

// ---------------------------------------------------------------- bias broadcast add
__global__ void add_bias(float* __restrict__ out, const float* __restrict__ b,
                         int NC, long total) {
    long i = (long)blockIdx.x * blockDim.x + threadIdx.x;
    if (i >= total) return;
    out[i] += b[(int)(i % NC)];
}

// ================================================================ host launch
static inline int cdiv(long a, int b) { return (int)((a + b - 1) / b); }

extern "C" void kernel_launch(void* const* d_in, const int* in_sizes, int n_in,
                              void* d_out, int out_size, void* d_ws, size_t ws_size,
                              hipStream_t stream) {
    const float* X        = (const float*)d_in[0];
    const int*   ei       = (const int*)  d_in[1];
    const float* W1       = (const float*)d_in[2];
    const float* att_src1 = (const float*)d_in[3];
    const float* att_dst1 = (const float*)d_in[4];
    const float* b1       = (const float*)d_in[5];
    const float* W2       = (const float*)d_in[6];
    const float* att_src2 = (const float*)d_in[7];
    const float* att_dst2 = (const float*)d_in[8];
    const float* b2       = (const float*)d_in[9];
    float* out = (float*)d_out;

    // ---- workspace layout (floats); big buffers reused across layers ----
    float* ws = (float*)d_ws;
    long off = 0;
    float*    xh1   = ws + off; off += (long)NNODES * 256;   // layer1 GEMM out; reused as xh2
    float*    h1    = ws + off; off += (long)NNODES * 256;   // layer1 aggregate+bias (layer2 input)
    float*    esum1 = ws + off; off += (long)NNODES * 4;
    unsigned* emax1 = (unsigned*)(ws + off); off += (long)NNODES * 4;
    float*    esum2 = ws + off; off += NNODES;
    unsigned* emax2 = (unsigned*)(ws + off); off += NNODES;
    float*    asrc1 = ws + off; off += (long)NNODES * 4;     // reused as asrc2
    float*    adst1 = ws + off; off += (long)NNODES * 4;     // reused as adst2
    float*    e1    = ws + off; off += (long)ETOT * 4;       // reused as e2
    float*    xh2   = xh1;
    float*    e2    = e1;
    float*    asrc2 = asrc1;
    float*    adst2 = adst1;

    const int B = 256;

    // zero the accumulator / max / sum regions (h1..emax2 are contiguous) and d_out
    long zlen = (long)NNODES * 256 + (long)NNODES * 4 * 2 + (long)NNODES * 2;
    zero_f32<<<2048, B, 0, stream>>>(h1, zlen);
    zero_f32<<<2048, B, 0, stream>>>(out, (long)NNODES * 128);

    // ---------------- Layer 1: heads=4, C=64, concat ----------------
    {
        int waves = (NNODES / 16) * (256 / 64);                       // 12500
        gemm_f32_wmma<<<cdiv((long)waves * 32, B), B, 0, stream>>>(X, W1, xh1, NNODES, 256, 64);
        attn_scores<<<cdiv((long)NNODES * 4, B), B, 0, stream>>>(xh1, att_src1, att_dst1,
                                                                 asrc1, adst1, 4, 64);
        edge_max   <<<cdiv((long)ETOT * 4, B), B, 0, stream>>>(ei, asrc1, adst1, e1, emax1, 4);
        edge_expsum<<<cdiv((long)ETOT * 4, B), B, 0, stream>>>(ei, e1, emax1, esum1, 4);
        edge_aggregate<<<cdiv((long)ETOT * 32, B), B, 0, stream>>>(ei, e1, esum1, xh1, h1, 4, 64);
        add_bias<<<cdiv((long)NNODES * 256, B), B, 0, stream>>>(h1, b1, 256, (long)NNODES * 256);
    }

    // ---------------- Layer 2: heads=1, C=128 (mean over 1 head == identity) ----------------
    {
        int waves = (NNODES / 16) * (128 / 64);                       // 6250
        gemm_f32_wmma<<<cdiv((long)waves * 32, B), B, 0, stream>>>(h1, W2, xh2, NNODES, 128, 256);
        attn_scores<<<cdiv((long)NNODES, B), B, 0, stream>>>(xh2, att_src2, att_dst2,
                                                             asrc2, adst2, 1, 128);
        edge_max   <<<cdiv((long)ETOT, B), B, 0, stream>>>(ei, asrc2, adst2, e2, emax2, 1);
        edge_expsum<<<cdiv((long)ETOT, B), B, 0, stream>>>(ei, e2, emax2, esum2, 1);
        edge_aggregate<<<cdiv((long)ETOT * 32, B), B, 0, stream>>>(ei, e2, esum2, xh2, out, 1, 128);
        add_bias<<<cdiv((long)NNODES * 128, B), B, 0, stream>>>(out, b2, 128, (long)NNODES * 128);
    }
}